// InverseWaveletBlock2D_64415919505834
// MI455X (gfx1250) — compile-verified
//
#include <hip/hip_runtime.h>
#include <hip/hip_bf16.h>

// ---------------------------------------------------------------------------
// 3-level inverse 2-D wavelet synthesis (K=10 taps, P=2 antireflect pad),
// fp32 end-to-end. Each 16x16 output tile of a 1-D polyphase synthesis pass
// is a pair of 16x16x12 matmuls on the CDNA5 matrix pipe, built from three
// chained V_WMMA_F32_16X16X4_F32 per operand matrix (the synthesis band
// matrix Hm[o,w] is zero for w >= 12, so K=12 suffices):
//
//   column pass:  Y(16x16) = Hm(16x12) * Awin(12x16) + Gm * Dwin
//   row pass:     Y(16x16) = Swin(16x12) * Hm^T      + Dwin * Gm^T
//
// Hm[o,w] = hk[o&1, w-(o>>1)] for 0 <= w-(o>>1) < 5, else 0, where
// hk[p,t] = h[8-2t+p] (flipped polyphase decomposition of h).
//
// Only the first/last tile along the transform axis can touch the pad; that
// check is wave-uniform. Interior tiles use straight coalesced loads with
// compile-time strides; boundary tiles use a *branchless* antireflect fetch
// (index select + 2 loads + blend, no EXEC divergence). One wave per tile,
// 8 waves per 256-thread block (wave32).
// ---------------------------------------------------------------------------

typedef float v2f __attribute__((ext_vector_type(2)));
typedef float v8f __attribute__((ext_vector_type(8)));

// Branchless antireflect ('reflect', reflect_type='odd') fetch, generic axis.
// i in [-2, N+1] by construction. Result: interior -> x[i];
// left pad -> 2*x[0] - x[-i]; right pad -> 2*x[N-1] - x[2N-2-i].
__device__ __forceinline__ float fetch_refl(const float* __restrict__ img,
                                            int stride, int N, int i, int off) {
  const bool lo = (i < 0);
  const bool hi = (i >= N);
  const int ref  = lo ? -i : (hi ? 2 * N - 2 - i : i);
  const int edge = hi ? N - 1 : 0;
  const float xr = img[(size_t)ref * stride + off];
  const float xe = img[(size_t)edge * stride + off];
  return (lo || hi) ? __builtin_fmaf(xe, 2.0f, -xr) : xr;
}

// Per-lane filter fragment, chunk c in [0,3), vgpr v: element Hm[m, 4c+2*half+v].
// Serves as the WMMA-A fragment in the column pass and (identically) as the
// Hm^T WMMA-B fragment in the row pass.
__device__ __forceinline__ void filter_frag(const float* __restrict__ h,
                                            const float* __restrict__ g,
                                            int m, int half, v2f HF[3], v2f GF[3]) {
  const int p = m & 1;
  const int n = m >> 1;
#pragma unroll
  for (int c = 0; c < 3; ++c) {
#pragma unroll
    for (int v = 0; v < 2; ++v) {
      const int k = 4 * c + 2 * half + v;
      const int t = k - n;
      const bool in = (t >= 0) && (t < 5);
      HF[c][v] = in ? h[8 - 2 * t + p] : 0.0f;
      GF[c][v] = in ? g[8 - 2 * t + p] : 0.0f;
    }
  }
}

// ---------------------------------------------------------------------------
// Column-synthesis pass: a,d are [B, N, W]; out is [B, 2N, W].
// out[2n+p, c] = sum_t hk[p,t] a_pad[n+t, c] + gk[p,t] d_pad[n+t, c]
// ---------------------------------------------------------------------------
template <int N, int W>
__global__ __launch_bounds__(256)
void idwt_col_wmma(const float* __restrict__ a, const float* __restrict__ d,
                   float* __restrict__ out, const float* __restrict__ h,
                   const float* __restrict__ g) {
  const int lane = threadIdx.x & 31;
  const int wave = threadIdx.x >> 5;
  const int m    = lane & 15;
  const int half = lane >> 4;

  const int c0  = (blockIdx.x * 8 + wave) * 16;  // column tile
  const int ty  = blockIdx.y;                    // output-row tile (2N/16 tiles)
  const int n0  = ty * 8;                        // input-row base
  const int col = c0 + m;
  const int b   = blockIdx.z;

  const float* __restrict__ A = a + (size_t)b * N * W;
  const float* __restrict__ D = d + (size_t)b * N * W;
  float*       __restrict__ O = out + (size_t)b * (2 * N) * W;

  v2f HF[3], GF[3];
  filter_frag(h, g, m, half, HF, GF);

  v2f Ab[3], Db[3];
  const bool interior = (ty > 0) && (ty < (N / 8) - 1);  // wave-uniform
  if (interior) {
    const float* __restrict__ Ap = A + (size_t)(n0 - 2 + 2 * half) * W + col;
    const float* __restrict__ Dp = D + (size_t)(n0 - 2 + 2 * half) * W + col;
#pragma unroll
    for (int c = 0; c < 3; ++c) {
      Ab[c][0] = Ap[(size_t)(4 * c) * W];
      Ab[c][1] = Ap[(size_t)(4 * c + 1) * W];
      Db[c][0] = Dp[(size_t)(4 * c) * W];
      Db[c][1] = Dp[(size_t)(4 * c + 1) * W];
    }
  } else {
#pragma unroll
    for (int c = 0; c < 3; ++c) {
#pragma unroll
      for (int v = 0; v < 2; ++v) {
        const int i = n0 + 4 * c + 2 * half + v - 2;  // a_pad[n0+k] = a[n0+k-P]
        Ab[c][v] = fetch_refl(A, W, N, i, col);
        Db[c][v] = fetch_refl(D, W, N, i, col);
      }
    }
  }

  v8f acc = {};
#pragma unroll
  for (int c = 0; c < 3; ++c) {
    acc = __builtin_amdgcn_wmma_f32_16x16x4_f32(false, HF[c], false, Ab[c],
                                                (short)0, acc, false, false);
    acc = __builtin_amdgcn_wmma_f32_16x16x4_f32(false, GF[c], false, Db[c],
                                                (short)0, acc, false, false);
  }

  const int o0 = ty * 16;
#pragma unroll
  for (int v = 0; v < 8; ++v) {
    const int M = v + 8 * half;                   // C/D layout: row = v + 8*half
    O[(size_t)(o0 + M) * W + col] = acc[v];
  }
}

// ---------------------------------------------------------------------------
// Row-synthesis pass: s,d are [B, H, Ws]; out is [B, H, 2*Ws].
// out[r, 2n+p] = sum_t hk[p,t] s_pad[r, n+t] + gk[p,t] d_pad[r, n+t]
// STREAM_OUT uses non-temporal stores (final output is write-once).
// ---------------------------------------------------------------------------
template <int H, int Ws, bool STREAM_OUT>
__global__ __launch_bounds__(256)
void idwt_row_wmma(const float* __restrict__ s, const float* __restrict__ d,
                   float* __restrict__ out, const float* __restrict__ h,
                   const float* __restrict__ g) {
  const int lane = threadIdx.x & 31;
  const int wave = threadIdx.x >> 5;
  const int m    = lane & 15;
  const int half = lane >> 4;

  const int tx = blockIdx.x * 8 + wave;          // output-col tile (2*Ws/16)
  const int o0 = tx * 16;
  const int r0 = blockIdx.y * 16;
  const int n0 = tx * 8;
  const int b  = blockIdx.z;
  constexpr int Wout = 2 * Ws;

  const float* __restrict__ S = s + (size_t)b * H * Ws;
  const float* __restrict__ D = d + (size_t)b * H * Ws;
  float*       __restrict__ O = out + (size_t)b * H * Wout;

  v2f HF[3], GF[3];                              // = Hm^T / Gm^T B-fragments
  filter_frag(h, g, m, half, HF, GF);

  const float* __restrict__ srow = S + (size_t)(r0 + m) * Ws;
  const float* __restrict__ drow = D + (size_t)(r0 + m) * Ws;

  v2f Sa[3], Da[3];
  const bool interior = (tx > 0) && (tx < (Ws / 8) - 1);  // wave-uniform
  if (interior) {
    const int base = n0 - 2 + 2 * half;          // even -> 8B aligned
#pragma unroll
    for (int c = 0; c < 3; ++c) {
      Sa[c] = *(const v2f*)(srow + base + 4 * c);  // global_load_b64
      Da[c] = *(const v2f*)(drow + base + 4 * c);
    }
  } else {
#pragma unroll
    for (int c = 0; c < 3; ++c) {
#pragma unroll
      for (int v = 0; v < 2; ++v) {
        const int j = n0 + 4 * c + 2 * half + v - 2;
        Sa[c][v] = fetch_refl(srow, 1, Ws, j, 0);
        Da[c][v] = fetch_refl(drow, 1, Ws, j, 0);
      }
    }
  }

  v8f acc = {};
#pragma unroll
  for (int c = 0; c < 3; ++c) {
    acc = __builtin_amdgcn_wmma_f32_16x16x4_f32(false, Sa[c], false, HF[c],
                                                (short)0, acc, false, false);
    acc = __builtin_amdgcn_wmma_f32_16x16x4_f32(false, Da[c], false, GF[c],
                                                (short)0, acc, false, false);
  }

#pragma unroll
  for (int v = 0; v < 8; ++v) {
    const int M = v + 8 * half;                  // row within tile
    float* p = O + (size_t)(r0 + M) * Wout + (o0 + m);
    if (STREAM_OUT) __builtin_nontemporal_store(acc[v], p);
    else            *p = acc[v];
  }
}

// ---------------------------------------------------------------------------
// Host orchestration. Inputs (setup_inputs order):
//   0:ss 1:sd0 2:sd1 3:sd2 4:ds0 5:ds1 6:ds2 7:dd0 8:dd1 9:dd2 10:h 11:g
// Workspace: two ping-pong images of 8*1024*512 fp32 (32 MB total).
// Intermediate-level ss outputs are staged inside d_out (consumed before the
// final pass overwrites all of d_out). All launches on `stream`; deterministic.
// ---------------------------------------------------------------------------
template <int N, int W>
static inline void launch_col(const float* a, const float* d, float* o,
                              const float* h, const float* g, hipStream_t s) {
  dim3 grid((W / 16) / 8, (2 * N) / 16, 8);
  idwt_col_wmma<N, W><<<grid, dim3(256), 0, s>>>(a, d, o, h, g);
}

template <int H, int Ws, bool STREAM_OUT>
static inline void launch_row(const float* a, const float* d, float* o,
                              const float* h, const float* g, hipStream_t s) {
  dim3 grid((2 * Ws / 16) / 8, H / 16, 8);
  idwt_row_wmma<H, Ws, STREAM_OUT><<<grid, dim3(256), 0, s>>>(a, d, o, h, g);
}

extern "C" void kernel_launch(void* const* d_in, const int* in_sizes, int n_in,
                              void* d_out, int out_size, void* d_ws, size_t ws_size,
                              hipStream_t stream) {
  (void)in_sizes; (void)n_in; (void)out_size; (void)ws_size;

  const float* ss  = (const float*)d_in[0];
  const float* sd0 = (const float*)d_in[1];
  const float* sd1 = (const float*)d_in[2];
  const float* sd2 = (const float*)d_in[3];
  const float* ds0 = (const float*)d_in[4];
  const float* ds1 = (const float*)d_in[5];
  const float* ds2 = (const float*)d_in[6];
  const float* dd0 = (const float*)d_in[7];
  const float* dd1 = (const float*)d_in[8];
  const float* dd2 = (const float*)d_in[9];
  const float* h   = (const float*)d_in[10];
  const float* g   = (const float*)d_in[11];

  float* out  = (float*)d_out;
  float* bufS = (float*)d_ws;
  float* bufD = bufS + (size_t)8 * 1024 * 512;
  float* bufO = out;                      // scratch for intermediate ss levels

  // ---- level 2: 128 -> 256 ----
  launch_col<128, 128>(ss,  sd2, bufS, h, g, stream);
  launch_col<128, 128>(ds2, dd2, bufD, h, g, stream);
  launch_row<256, 128, false>(bufS, bufD, bufO, h, g, stream);

  // ---- level 1: 256 -> 512 ----
  launch_col<256, 256>(bufO, sd1, bufS, h, g, stream);
  launch_col<256, 256>(ds1,  dd1, bufD, h, g, stream);
  launch_row<512, 256, false>(bufS, bufD, bufO, h, g, stream);

  // ---- level 0: 512 -> 1024 (final output, streamed) ----
  launch_col<512, 512>(bufO, sd0, bufS, h, g, stream);
  launch_col<512, 512>(ds0,  dd0, bufD, h, g, stream);
  launch_row<1024, 512, true>(bufS, bufD, out, h, g, stream);
}